// SparseAutoEncoder_37666863186493
// MI455X (gfx1250) — compile-verified
//
#include <hip/hip_runtime.h>
#include <hip/hip_bf16.h>

typedef __attribute__((ext_vector_type(2))) float v2f;
typedef __attribute__((ext_vector_type(4))) float v4f;
typedef __attribute__((ext_vector_type(8))) float v8f;

#define SEQ_LEN      128
#define SAE_HIDDEN   2048
#define RES_DIM      768

// ---------------------------------------------------------------------------
// Kernel A: transpose W_dec [768, 2048] -> Wt [2048, 768] (6.3 MB, one-shot)
// so that Wt rows (fixed h, d contiguous) are coalesced for the GEMM + stream.
// ---------------------------------------------------------------------------
__global__ void sae_transpose_wdec(const float* __restrict__ wdec,
                                   float* __restrict__ wt) {
    __shared__ float tile[32][33];
    int x = blockIdx.x * 32 + threadIdx.x;   // src col (h), 0..2047
    int y = blockIdx.y * 32 + threadIdx.y;   // src row (d), 0..767
    #pragma unroll
    for (int j = 0; j < 32; j += 8)
        tile[threadIdx.y + j][threadIdx.x] = wdec[(y + j) * SAE_HIDDEN + x];
    __syncthreads();
    int xo = blockIdx.y * 32 + threadIdx.x;  // dst col (d), 0..767
    int yo = blockIdx.x * 32 + threadIdx.y;  // dst row (h), 0..2047
    #pragma unroll
    for (int j = 0; j < 32; j += 8)
        wt[(yo + j) * RES_DIM + xo] = tile[threadIdx.x][threadIdx.y + j];
}

// ---------------------------------------------------------------------------
// Kernel B: encoder GEMM via V_WMMA_F32_16X16X4_F32
//   sae[s,h] = relu( sum_d X[s,d] * W_enc[h,d] + b_enc[h] )
//   M = 128 (s), N = 2048 (h), K = 768 (d). One 16x16 tile per wave32.
// A frag (16x4 f32): lane<16 holds row M=lane, K = k,k+1; lane>=16 K = k+2,k+3
// B frag (4x16 f32): lane<16 holds col N=lane, K = k,k+1; lane>=16 K = k+2,k+3
// C/D (16x16 f32, 8 VGPR): vgpr v, lane L -> M = v + 8*(L>=16), N = L%16
// ---------------------------------------------------------------------------
__global__ void sae_encoder_wmma(const float* __restrict__ X,
                                 const float* __restrict__ Wenc,
                                 const float* __restrict__ benc,
                                 float* __restrict__ sae) {
    const int lane = threadIdx.x & 31;
    const int wave = threadIdx.x >> 5;
    const int gw   = blockIdx.x * 4 + wave;          // 1024 waves total
    const int tile_m = gw >> 7;                      // 0..7   (128/16)
    const int tile_n = gw & 127;                     // 0..127 (2048/16)
    const int m0 = tile_m * 16;
    const int n0 = tile_n * 16;

    const int half = lane >> 4;                      // 0 | 1
    const int l16  = lane & 15;
    const int arow = m0 + l16;                       // A row (s)
    const int bcol = n0 + l16;                       // B col (h)

    v8f c = {};
    for (int k = 0; k < RES_DIM; k += 4) {
        const int kb = k + 2 * half;
        v2f a = *(const v2f*)(X    + arow * RES_DIM + kb);
        v2f b = *(const v2f*)(Wenc + bcol * RES_DIM + kb);
        c = __builtin_amdgcn_wmma_f32_16x16x4_f32(false, a, false, b,
                                                  (short)0, c, false, false);
    }

    const float bias = benc[bcol];                   // column bias (per-N)
    #pragma unroll
    for (int v = 0; v < 8; ++v) {
        float r = c[v] + bias;
        r = fmaxf(r, 0.0f);                          // ReLU
        sae[(m0 + v + 8 * half) * SAE_HIDDEN + (n0 + l16)] = r;
    }
}

// ---------------------------------------------------------------------------
// Kernel C: decoder GEMM via V_WMMA_F32_16X16X4_F32
//   rec[s,d] = sum_h sae[s,h] * Wt[h,d] + b_dec[d]
//   M = 128 (s), N = 768 (d), K = 2048 (h).
// ---------------------------------------------------------------------------
__global__ void sae_decoder_wmma(const float* __restrict__ sae,
                                 const float* __restrict__ Wt,
                                 const float* __restrict__ bdec,
                                 float* __restrict__ rec) {
    const int lane = threadIdx.x & 31;
    const int wave = threadIdx.x >> 5;
    const int gw   = blockIdx.x * 4 + wave;          // 384 waves total
    const int tile_m = gw / 48;                      // 0..7  (128/16)
    const int tile_n = gw % 48;                      // 0..47 (768/16)
    const int m0 = tile_m * 16;
    const int n0 = tile_n * 16;

    const int half = lane >> 4;
    const int l16  = lane & 15;
    const int arow = m0 + l16;                       // sae row (s)
    const int bcol = n0 + l16;                       // d

    v8f c = {};
    for (int k = 0; k < SAE_HIDDEN; k += 4) {
        const int kb = k + 2 * half;
        v2f a = *(const v2f*)(sae + arow * SAE_HIDDEN + kb);
        v2f b;
        b.x = Wt[(kb    ) * RES_DIM + bcol];
        b.y = Wt[(kb + 1) * RES_DIM + bcol];
        c = __builtin_amdgcn_wmma_f32_16x16x4_f32(false, a, false, b,
                                                  (short)0, c, false, false);
    }

    const float bias = bdec[bcol];
    #pragma unroll
    for (int v = 0; v < 8; ++v)
        rec[(m0 + v + 8 * half) * RES_DIM + (n0 + l16)] = c[v] + bias;
}

// ---------------------------------------------------------------------------
// Kernel D: feat_vecs streamer (the 805 MB store stream — the whole problem).
//   out[s,h,d] = sae[s,h] * Wt[h,d]
// One block per (s,h) row: 192 threads x float4 = 768 floats.
// sae[hs] is block-uniform -> scalar load broadcast. Wt row read is coalesced
// and L2-resident (6.3 MB << 192 MB). Stores are non-temporal b128 so the
// write stream does not evict Wt from L2.
// ---------------------------------------------------------------------------
__global__ void sae_featvec_stream(const float* __restrict__ sae,
                                   const float* __restrict__ Wt,
                                   float* __restrict__ out) {
    const int hs = blockIdx.x;                       // s*2048 + h, 0..262143
    const int h  = hs & (SAE_HIDDEN - 1);
    const int d4 = threadIdx.x;                      // 0..191 (float4 lanes)

    const float a = sae[hs];                         // uniform within block
    const v4f  w  = ((const v4f*)Wt)[h * (RES_DIM / 4) + d4];
    const v4f  r  = a * w;
    __builtin_nontemporal_store(r,
        (v4f*)out + (long)hs * (RES_DIM / 4) + d4);
}

// ---------------------------------------------------------------------------
extern "C" void kernel_launch(void* const* d_in, const int* in_sizes, int n_in,
                              void* d_out, int out_size, void* d_ws, size_t ws_size,
                              hipStream_t stream) {
    const float* X    = (const float*)d_in[0];   // [1,128,768]
    const float* Wenc = (const float*)d_in[1];   // [2048,768]
    const float* benc = (const float*)d_in[2];   // [2048]
    const float* Wdec = (const float*)d_in[3];   // [768,2048]
    const float* bdec = (const float*)d_in[4];   // [768]

    float* rec  = (float*)d_out;                         // [128,768]
    float* feat = rec + SEQ_LEN * RES_DIM;               // [128,2048,768]

    float* sae = (float*)d_ws;                           // 128*2048 f32 (1 MB)
    float* Wt  = sae + SEQ_LEN * SAE_HIDDEN;             // 2048*768 f32 (6.3 MB)

    // A: transpose W_dec -> Wt
    sae_transpose_wdec<<<dim3(SAE_HIDDEN / 32, RES_DIM / 32), dim3(32, 8), 0, stream>>>(Wdec, Wt);

    // B: encoder GEMM + bias + ReLU -> sae (1024 waves, 4 waves/block)
    sae_encoder_wmma<<<256, 128, 0, stream>>>(X, Wenc, benc, sae);

    // C: decoder GEMM + bias -> reconstructed (384 waves)
    sae_decoder_wmma<<<96, 128, 0, stream>>>(sae, Wt, bdec, rec);

    // D: feat_vecs stream: 262,144 blocks x 192 threads, NT b128 stores
    sae_featvec_stream<<<SEQ_LEN * SAE_HIDDEN, RES_DIM / 4, 0, stream>>>(sae, Wt, feat);
}